// QuadAE_16458314678353
// MI455X (gfx1250) — compile-verified
//
#include <hip/hip_runtime.h>

// CDNA5 / gfx1250 — wave32, WMMA f32 16x16x4, LDS-staged weights, unrolled K.
typedef __attribute__((ext_vector_type(2))) float v2f;
typedef __attribute__((ext_vector_type(8))) float v8f;

#define N3_ 262144
#define N2_ 65536
#define N1_ 16384

__device__ __forceinline__ int lower_bound_i32(const int* __restrict__ a, int n, int key) {
  int lo = 0, hi = n;
  while (lo < hi) { int mid = (lo + hi) >> 1; if (a[mid] < key) lo = mid + 1; else hi = mid; }
  return lo;  // == jnp.searchsorted(a, key) (side='left')
}

// ---------------------------------------------------------------------------
// Quadconv as WMMA GEMM: out[n,o] = relu?( b[o] + sum_k col[n,k]*w[o,k] )
// col[n, j*CIN+c] = (neigh[n][j] < 0) ? 0 : feat[neigh[n][j]][c]
// Block = 8 waves; wave w owns row tile (blockIdx.x*8+w); block stages its
// 16*COLT weight rows into LDS once, inner loop reads B via ds_load_b64.
// A frag (16x4 f32): lane m=lane&15 -> row m; half=lane>>4 selects K pair
//   (VGPR0 = K0/K2, VGPR1 = K1/K3) -> one 8B global load, immediate offset.
// C/D frag: VGPR r -> (M = r + 8*half, N = lane&15).  A reused across COLT.
// ---------------------------------------------------------------------------
template <int CIN, int COLT>
__global__ __launch_bounds__(256)
void quadconv_wmma(const float* __restrict__ feat, const int* __restrict__ neigh,
                   const float* __restrict__ w, const float* __restrict__ bias,
                   float* __restrict__ out, int Cout, int doRelu) {
  constexpr int K     = 9 * CIN;
  constexpr int WTILE = 16 * COLT;
  constexpr int WSZ   = (CIN >= 4) ? WTILE * K : 4;   // LDS floats (dummy for CIN==1)
  __shared__ float wlds[WSZ];

  const int lane = threadIdx.x & 31;
  const int wave = threadIdx.x >> 5;
  const int m    = lane & 15;
  const int half = lane >> 4;
  const long rowBase = ((long)blockIdx.x * 8 + wave) * 16;
  const int colBase  = blockIdx.y * WTILE;

  if constexpr (CIN >= 4) {
    // Cooperative stage of w[colBase .. colBase+WTILE) rows into LDS (b128s).
    const int validRows = (Cout - colBase) < WTILE ? (Cout - colBase) : WTILE;
    const int v4 = (validRows * K) >> 2;
    const float4* wg = (const float4*)(w + (long)colBase * K);
    for (int e = threadIdx.x; e < v4; e += 256) ((float4*)wlds)[e] = wg[e];
    __syncthreads();
  }

  int nb[9];
#pragma unroll
  for (int j = 0; j < 9; ++j) nb[j] = neigh[(rowBase + m) * 9 + j];

  int  oc[COLT]; bool oOK[COLT];
#pragma unroll
  for (int t = 0; t < COLT; ++t) {
    oc[t]  = colBase + 16 * t + m;
    oOK[t] = (oc[t] < Cout);
  }

  v8f acc[COLT];
#pragma unroll
  for (int t = 0; t < COLT; ++t) acc[t] = (v8f){0.f,0.f,0.f,0.f,0.f,0.f,0.f,0.f};

  if constexpr (CIN == 1) {
    // K = 9: materialize im2col row + weight row in registers (pad to 12);
    // constant-index with a cndmask on `half` per unrolled step.
    float colv[12], wv[12];
#pragma unroll
    for (int j = 0; j < 9; ++j) {
      const int idx = nb[j];
      colv[j] = (idx >= 0) ? feat[idx] : 0.f;
      wv[j]   = oOK[0] ? w[(long)oc[0] * K + j] : 0.f;
    }
    colv[9] = colv[10] = colv[11] = 0.f;
    wv[9]   = wv[10]   = wv[11]   = 0.f;
#pragma unroll
    for (int k0 = 0; k0 < 12; k0 += 4) {
      v2f A = { half ? colv[k0 + 2] : colv[k0 + 0],
                half ? colv[k0 + 3] : colv[k0 + 1] };
      v2f B = { half ? wv[k0 + 2] : wv[k0 + 0],
                half ? wv[k0 + 3] : wv[k0 + 1] };
      acc[0] = __builtin_amdgcn_wmma_f32_16x16x4_f32(false, A, false, B,
                                                     (short)0, acc[0], false, false);
    }
  } else {
    // Per-neighbor base pointers incl. half-wave K-pair offset; all accesses
    // are 8B loads at compile-time immediate offsets (global A, LDS B).
    const float* ap[9]; bool aok[9];
#pragma unroll
    for (int j = 0; j < 9; ++j) {
      const int idx = nb[j];
      aok[j] = (idx >= 0);
      ap[j]  = feat + (long)(idx >= 0 ? idx : 0) * CIN + 2 * half;
    }
    const float* wlp[COLT];
#pragma unroll
    for (int t = 0; t < COLT; ++t)
      wlp[t] = wlds + (oOK[t] ? (16 * t + m) * K : 0) + 2 * half;
#pragma unroll
    for (int j = 0; j < 9; ++j) {
#pragma unroll
      for (int c0 = 0; c0 < CIN; c0 += 4) {
        v2f A = *(const v2f*)(ap[j] + c0);
        if (!aok[j]) A = (v2f){0.f, 0.f};
#pragma unroll
        for (int t = 0; t < COLT; ++t) {
          v2f B = *(const v2f*)(wlp[t] + j * CIN + c0);   // ds_load_b64
          acc[t] = __builtin_amdgcn_wmma_f32_16x16x4_f32(false, A, false, B,
                                                         (short)0, acc[t], false, false);
        }
      }
    }
  }

#pragma unroll
  for (int t = 0; t < COLT; ++t) {
    if (!oOK[t]) continue;
    const float bv = bias[oc[t]];
#pragma unroll
    for (int r = 0; r < 8; ++r) {
      const long orow = rowBase + r + half * 8;
      float v = acc[t][r] + bv;
      if (doRelu) v = v > 0.f ? v : 0.f;
      out[orow * Cout + oc[t]] = v;
    }
  }
}

// Output-centric pool: searchsorted(keysP, keysC[i]>>2) == p  <=>
// keysC[i] in [ (keysP[p-1]+1)<<2 , (keysP[p]+1)<<2 )  (keys sorted).
// Inputs are post-ReLU (>= 0), so zero-init == reference's -inf -> 0 fill.
__global__ void quadpool_seg(const float* __restrict__ x, const int* __restrict__ keysC,
                             const int* __restrict__ keysP, float* __restrict__ out,
                             int Nc, int Np, int C) {
  int p = blockIdx.x * blockDim.x + threadIdx.x;
  if (p >= Np) return;
  const int start = (p == 0) ? 0
                  : lower_bound_i32(keysC, Nc, (keysP[p - 1] + 1) << 2);
  const int end   = lower_bound_i32(keysC, Nc, (keysP[p] + 1) << 2);
  for (int c4 = 0; c4 < (C >> 2); ++c4) {
    float4 acc = {0.f, 0.f, 0.f, 0.f};
    for (int i = start; i < end; ++i) {
      float4 v = *(const float4*)(x + (long)i * C + c4 * 4);
      acc.x = fmaxf(acc.x, v.x); acc.y = fmaxf(acc.y, v.y);
      acc.z = fmaxf(acc.z, v.z); acc.w = fmaxf(acc.w, v.w);
    }
    *(float4*)(out + (long)p * C + c4 * 4) = acc;
  }
}

// Unpool: out[i] = feat[searchsorted(parent_keys, key>>2)] (JAX clamps OOB gather).
__global__ void quadunpool(const float* __restrict__ x, const int* __restrict__ keysC,
                           const int* __restrict__ keysP, float* __restrict__ out,
                           int Nc, int Np, int C) {
  int i = blockIdx.x * blockDim.x + threadIdx.x;
  if (i >= Nc) return;
  int p = lower_bound_i32(keysP, Np, keysC[i] >> 2);
  if (p >= Np) p = Np - 1;
  const float4* src = (const float4*)(x + (long)p * C);
  float4*       dst = (float4*)(out + (long)i * C);
  for (int c = 0; c < (C >> 2); ++c) dst[c] = src[c];
}

extern "C" void kernel_launch(void* const* d_in, const int* in_sizes, int n_in,
                              void* d_out, int out_size, void* d_ws, size_t ws_size,
                              hipStream_t stream) {
  (void)in_sizes; (void)n_in; (void)out_size; (void)ws_size;
  const float* features = (const float*)d_in[0];
  const int*   keys3    = (const int*)d_in[1];
  const int*   keys2    = (const int*)d_in[2];
  const int*   keys1    = (const int*)d_in[3];
  const int*   neighs3  = (const int*)d_in[4];
  const int*   neighs2  = (const int*)d_in[5];
  const int*   neighs1  = (const int*)d_in[6];
  const float* w_enc1 = (const float*)d_in[7];  const float* b_enc1 = (const float*)d_in[8];
  const float* w_enc2 = (const float*)d_in[9];  const float* b_enc2 = (const float*)d_in[10];
  const float* w_enc3 = (const float*)d_in[11]; const float* b_enc3 = (const float*)d_in[12];
  const float* w_dec1 = (const float*)d_in[13]; const float* b_dec1 = (const float*)d_in[14];
  const float* w_dec2 = (const float*)d_in[15]; const float* b_dec2 = (const float*)d_in[16];
  const float* w_head = (const float*)d_in[17]; const float* b_head = (const float*)d_in[18];

  // Workspace (float offsets); lifetime-based reuse, peak 15M floats = 60 MB.
  float* ws = (float*)d_ws;
  const long MEG = 1024 * 1024;
  float* x8    = ws;                      // [N3,16]  0 .. 4M      (dead after pool1)
  float* x7p   = ws + 4 * MEG;            // [N2,16]  4M .. 5M
  float* x7    = ws + 5 * MEG;            // [N2,32]  5M .. 7M     (dead after pool2)
  float* x6p   = ws + 7 * MEG;            // [N1,32]  7M .. 7.5M
  float* x6    = ws + 7 * MEG + MEG / 2;  // [N1,64]  7.5M .. 8.5M
  float* x7up  = ws + 8 * MEG + MEG / 2;  // [N2,64]  8.5M .. 12.5M (dead after conv4)
  float* x7dec = ws + 5 * MEG;            // [N2,32]  reuse of x7 region
  float* x8up  = ws + 7 * MEG;            // [N3,32]  7M .. 15M (only x7dec live)
  float* x8dec = ws;                      // [N3,16]  reuse of x8 region

  const dim3 blk(256);

  // x8 = relu(conv(features, neighs3, w_enc1))           [N3,16], K=9
  quadconv_wmma<1, 1><<<dim3(N3_ / 128, 1), blk, 0, stream>>>(
      features, neighs3, w_enc1, b_enc1, x8, 16, 1);

  // x7_p = pool(x8)                                      [N2,16]
  quadpool_seg<<<(N2_ + 255) / 256, blk, 0, stream>>>(x8, keys3, keys2, x7p, N3_, N2_, 16);

  // x7 = relu(conv(x7_p, neighs2, w_enc2))               [N2,32], K=144
  quadconv_wmma<16, 2><<<dim3(N2_ / 128, 1), blk, 0, stream>>>(
      x7p, neighs2, w_enc2, b_enc2, x7, 32, 1);

  // x6_p = pool(x7)                                      [N1,32]
  quadpool_seg<<<(N1_ + 255) / 256, blk, 0, stream>>>(x7, keys2, keys1, x6p, N2_, N1_, 32);

  // x6 = relu(conv(x6_p, neighs1, w_enc3))               [N1,64], K=288
  quadconv_wmma<32, 2><<<dim3(N1_ / 128, 2), blk, 0, stream>>>(
      x6p, neighs1, w_enc3, b_enc3, x6, 64, 1);

  // x7_up = unpool(x6)                                   [N2,64]
  quadunpool<<<(N2_ + 255) / 256, blk, 0, stream>>>(x6, keys2, keys1, x7up, N2_, N1_, 64);

  // x7_dec = relu(conv(x7_up, neighs2, w_dec1))          [N2,32], K=576
  quadconv_wmma<64, 2><<<dim3(N2_ / 128, 1), blk, 0, stream>>>(
      x7up, neighs2, w_dec1, b_dec1, x7dec, 32, 1);

  // x8_up = unpool(x7_dec)                               [N3,32]
  quadunpool<<<(N3_ + 255) / 256, blk, 0, stream>>>(x7dec, keys3, keys2, x8up, N3_, N2_, 32);

  // x8_dec = relu(conv(x8_up, neighs3, w_dec2))          [N3,16], K=288
  quadconv_wmma<32, 1><<<dim3(N3_ / 128, 1), blk, 0, stream>>>(
      x8up, neighs3, w_dec2, b_dec2, x8dec, 16, 1);

  // out = conv(x8_dec, neighs3, w_head)  (no relu)       [N3,1], K=144
  quadconv_wmma<16, 1><<<dim3(N3_ / 128, 1), blk, 0, stream>>>(
      x8dec, neighs3, w_head, b_head, (float*)d_out, 1, 0);
}